// EuclideanGating_66314295050615
// MI455X (gfx1250) — compile-verified
//
#include <hip/hip_runtime.h>
#include <hip/hip_bf16.h>

typedef __attribute__((ext_vector_type(2))) float v2f;
typedef __attribute__((ext_vector_type(8))) float v8f;

#define FEAT 64

// ---------------------------------------------------------------------------
// WMMA GEMM: C[M x (NT*16)] = A[M x K] @ W[K x (NT*16)]  (+ bias if BIAS)
// One wave32 per 16-row tile; V_WMMA_F32_16X16X4_F32, K stepped by 4.
// A frag (16x4 f32): lanes 0-15 hold K={k0,k0+1}, lanes 16-31 hold K={k0+2,k0+3}
// B frag (4x16 f32): mirrored (lane n%16 holds column n, K split across halves)
// C/D (16x16 f32, 8 VGPRs): VGPR v -> row v + 8*(lane>=16), col = lane%16
// ---------------------------------------------------------------------------
template <int NT, bool BIAS>
__global__ __launch_bounds__(256) void gemm_wmma_f32(
    const float* __restrict__ A, const float* __restrict__ W,
    const float* __restrict__ bias, float* __restrict__ C, int M, int K) {
  const int N = NT * 16;
  const int wave = (blockIdx.x * blockDim.x + threadIdx.x) >> 5;
  const int lane = threadIdx.x & 31;
  const int row0 = wave * 16;
  if (row0 >= M) return;  // whole wave exits together: EXEC all-ones for WMMA

  const int mL = lane & 15;          // column-in-tile / row-in-tile index
  const int kh = (lane >> 4) * 2;    // K sub-offset for this half-wave

  v8f acc[NT] = {};

  for (int k0 = 0; k0 < K; k0 += 4) {
    const float* ap = A + (row0 + mL) * K + (k0 + kh);
    v2f a;
    a.x = ap[0];
    a.y = ap[1];
#pragma unroll
    for (int t = 0; t < NT; ++t) {
      const float* wp = W + (k0 + kh) * N + t * 16 + mL;
      v2f b;
      b.x = wp[0];
      b.y = wp[N];
      acc[t] = __builtin_amdgcn_wmma_f32_16x16x4_f32(
          /*neg_a=*/false, a, /*neg_b=*/false, b,
          /*c_mod=*/(short)0, acc[t], /*reuse_a=*/false, /*reuse_b=*/false);
    }
  }

  const int mOut = (lane >> 4) * 8;
#pragma unroll
  for (int t = 0; t < NT; ++t) {
    const float bv = BIAS ? bias[t * 16 + mL] : 0.0f;
#pragma unroll
    for (int v = 0; v < 8; ++v) {
      C[(row0 + mOut + v) * N + t * 16 + mL] = acc[t][v] + bv;
    }
  }
}

// ---------------------------------------------------------------------------
// Graph kernels
// ---------------------------------------------------------------------------
__global__ void k_fill(float* __restrict__ p, float v, int n) {
  int i = blockIdx.x * blockDim.x + threadIdx.x;
  if (i < n) p[i] = v;
}

__global__ void k_degree(const int* __restrict__ dst, float* __restrict__ deg,
                         int E) {
  int e = blockIdx.x * blockDim.x + threadIdx.x;
  if (e < E) atomicAdd(&deg[dst[e]], 1.0f);
}

__global__ void k_inv_sqrt(const float* __restrict__ deg,
                           float* __restrict__ inv, int n) {
  int i = blockIdx.x * blockDim.x + threadIdx.x;
  if (i < n) inv[i] = rsqrtf(deg[i]);  // deg >= 1 (self-loop), no zero guard
}

// One thread per (edge, feature): agg[dst] += hw[src] * inv[src]*inv[dst]
__global__ __launch_bounds__(256) void k_scatter(
    const int* __restrict__ src, const int* __restrict__ dst,
    const float* __restrict__ inv, const float* __restrict__ hw,
    float* __restrict__ agg, int E) {
  int idx = blockIdx.x * blockDim.x + threadIdx.x;
  if (idx >= E * FEAT) return;
  int e = idx >> 6;
  int f = idx & (FEAT - 1);
  int s = src[e];
  int d = dst[e];
  float w = inv[s] * inv[d];
  atomicAdd(&agg[d * FEAT + f], hw[s * FEAT + f] * w);
}

// out = relu(agg + hw * (1/deg)  + bias)   (self-loop weight = inv[i]^2)
__global__ void k_epilogue(const float* __restrict__ agg,
                           const float* __restrict__ hw,
                           const float* __restrict__ inv,
                           const float* __restrict__ bias,
                           float* __restrict__ out, int n) {
  int idx = blockIdx.x * blockDim.x + threadIdx.x;
  if (idx >= n * FEAT) return;
  int i = idx >> 6;
  int f = idx & (FEAT - 1);
  float iw = inv[i];
  float v = agg[idx] + hw[idx] * (iw * iw) + bias[f];
  out[idx] = fmaxf(v, 0.0f);
}

// ---------------------------------------------------------------------------
extern "C" void kernel_launch(void* const* d_in, const int* in_sizes, int n_in,
                              void* d_out, int out_size, void* d_ws,
                              size_t ws_size, hipStream_t stream) {
  const float* x = (const float*)d_in[0];
  const int* edge = (const int*)d_in[1];
  const float* W1 = (const float*)d_in[2];
  const float* b1 = (const float*)d_in[3];
  const float* W2 = (const float*)d_in[4];
  const float* b2 = (const float*)d_in[5];
  const float* Wc = (const float*)d_in[6];
  const float* bc = (const float*)d_in[7];
  float* out = (float*)d_out;

  const int n = in_sizes[0] / FEAT;      // 50000
  const int E = in_sizes[1] / 2;         // 800000
  const int* src = edge;                 // edge_index[0]
  const int* dst = edge + E;             // edge_index[1]

  float* ws = (float*)d_ws;
  float* hw = ws;                        // n*64 : x@W (pre-aggregation)
  float* agg = hw + n * FEAT;            // n*64 : edge-aggregated sum
  float* h = agg + n * FEAT;             // n*64 : layer output
  float* deg = h + n * FEAT;             // n    : degree (incl. self-loop)
  float* inv = deg + n;                  // n    : deg^-1/2

  const int TB = 256;
  const int gN = (n + TB - 1) / TB;
  const int gE = (E + TB - 1) / TB;
  const int gNF = (n * FEAT + TB - 1) / TB;
  const int gEF = (E * FEAT + TB - 1) / TB;
  const int gWaves = ((n + 15) / 16) * 32;          // one wave per 16-row tile
  const int gGemm = (gWaves + TB - 1) / TB;

  // --- degrees / symmetric normalization (shared by both layers) ---
  k_fill<<<gN, TB, 0, stream>>>(deg, 1.0f, n);      // self-loop contributes 1
  k_degree<<<gE, TB, 0, stream>>>(dst, deg, E);
  k_inv_sqrt<<<gN, TB, 0, stream>>>(deg, inv, n);

  // --- layer 1: h = relu(Â (x@W1) + b1) ---
  gemm_wmma_f32<4, false><<<gGemm, TB, 0, stream>>>(x, W1, nullptr, hw, n, FEAT);
  k_fill<<<gNF, TB, 0, stream>>>(agg, 0.0f, n * FEAT);
  k_scatter<<<gEF, TB, 0, stream>>>(src, dst, inv, hw, agg, E);
  k_epilogue<<<gNF, TB, 0, stream>>>(agg, hw, inv, b1, h, n);

  // --- layer 2: h = relu(Â (h@W2) + b2) ---
  gemm_wmma_f32<4, false><<<gGemm, TB, 0, stream>>>(h, W2, nullptr, hw, n, FEAT);
  k_fill<<<gNF, TB, 0, stream>>>(agg, 0.0f, n * FEAT);
  k_scatter<<<gEF, TB, 0, stream>>>(src, dst, inv, hw, agg, E);
  k_epilogue<<<gNF, TB, 0, stream>>>(agg, hw, inv, b2, h, n);

  // --- classifier: out = h @ Wc + bc  (N=16, one C tile per wave) ---
  gemm_wmma_f32<1, true><<<gGemm, TB, 0, stream>>>(h, Wc, bc, out, n, FEAT);
}